// FWModule_24223615549826
// MI455X (gfx1250) — compile-verified
//
#include <hip/hip_runtime.h>

// Problem constants (match reference)
#define BB 8
#define TT 4096
#define DD 1024
#define NC 32              // chunks along T
#define CT (TT / NC)       // 128 tokens per chunk
#define PSZ (BB * NC * DD) // elements per partial/base buffer (1 MiB each as f32)

typedef __attribute__((ext_vector_type(2))) float v2f;
typedef __attribute__((ext_vector_type(8))) float v8f;

// ---------------------------------------------------------------------------
// Pass 1: per-chunk column sums of grad_w / grad_b.
// One thread per (b, chunk, d/4): loops CT steps with float4 loads.
// grid = BB*NC blocks, 256 threads (8 waves). 2048 waves total.
// ---------------------------------------------------------------------------
__global__ void fw_partial_sums(const float* __restrict__ gw,
                                const float* __restrict__ gb,
                                float* __restrict__ pw,
                                float* __restrict__ pb) {
    const int blk = blockIdx.x;            // b*NC + c
    const int b   = blk / NC;
    const int c   = blk % NC;
    const int d0  = threadIdx.x * 4;

    size_t idx = ((size_t)b * TT + (size_t)c * CT) * DD + d0;
    float sw0 = 0.f, sw1 = 0.f, sw2 = 0.f, sw3 = 0.f;
    float sb0 = 0.f, sb1 = 0.f, sb2 = 0.f, sb3 = 0.f;

#pragma unroll 4
    for (int t = 0; t < CT; ++t, idx += DD) {
        if (t + 16 < CT) {
            __builtin_prefetch((const char*)(gw + idx + 16 * DD), 0, 0);
            __builtin_prefetch((const char*)(gb + idx + 16 * DD), 0, 0);
        }
        const float4 g = *(const float4*)(gw + idx);
        const float4 h = *(const float4*)(gb + idx);
        sw0 += g.x; sw1 += g.y; sw2 += g.z; sw3 += g.w;
        sb0 += h.x; sb1 += h.y; sb2 += h.z; sb3 += h.w;
    }
    const size_t o = (size_t)blk * DD + d0;
    *(float4*)(pw + o) = make_float4(sw0, sw1, sw2, sw3);
    *(float4*)(pb + o) = make_float4(sb0, sb1, sb2, sb3);
}

// ---------------------------------------------------------------------------
// Pass 2: cross-chunk exclusive scan via f32 WMMA triangular matmul.
// base[c,d] = sum_{c'<c} P[c',d]  ==  L (strict lower ones) x P.
// 32 chunks = two 16x16 tiles; carry between tiles produced by an
// all-ones-A WMMA (every row of D = column sums of low tile) used as the
// C operand of the high tile. All arithmetic exact f32 (L entries are 0/1).
// One wave per (batch, 16-wide d tile): grid = 8*64 = 512 waves.
// Fragment layouts per ISA 7.12.2:
//   A 16x4 f32 : lanes 0-15 M=lane {VGPR0=K0, VGPR1=K1}; lanes 16-31 {K2,K3}
//   B 4x16 f32 : lanes 0-15 N=lane {VGPR0=K0, VGPR1=K1}; lanes 16-31 {K2,K3}
//   C/D 16x16  : VGPR r -> M = r (lanes 0-15) / r+8 (lanes 16-31), N = lane%16
// ---------------------------------------------------------------------------
__global__ void fw_scan_partials(const float* __restrict__ pw,
                                 const float* __restrict__ pb,
                                 float* __restrict__ bw,
                                 float* __restrict__ bb) {
    const int lane = threadIdx.x;      // 0..31, full wave (EXEC all ones)
    const int hf   = lane >> 4;        // lane half selects K offset {0,2}
    const int nl   = lane & 15;        // M for A fragments, N for B/C/D
    const int bi   = blockIdx.x >> 6;  // batch
    const int dt   = blockIdx.x & 63;  // 16-wide d tile
    const int d    = dt * 16 + nl;

    // A fragments: strict-lower-triangular ones (exclusive scan) + all-ones
    v2f La[4], Oa[4];
#pragma unroll
    for (int kk = 0; kk < 4; ++kk) {
        const int k0 = 4 * kk + 2 * hf;
        La[kk].x = (k0     < nl) ? 1.f : 0.f;
        La[kk].y = (k0 + 1 < nl) ? 1.f : 0.f;
        Oa[kk].x = 1.f;
        Oa[kk].y = 1.f;
    }

    const float* src[2] = {pw, pb};
    float*       dst[2] = {bw, bb};

    for (int mtx = 0; mtx < 2; ++mtx) {
        const float* __restrict__ P = src[mtx];
        float* __restrict__       O = dst[mtx];

        // B fragments: low tile = chunks 0..15, high tile = chunks 16..31
        v2f Blo[4], Bhi[4];
#pragma unroll
        for (int kk = 0; kk < 4; ++kk) {
            const int k0 = 4 * kk + 2 * hf;
            Blo[kk].x = P[((size_t)(bi * NC + k0     ) * DD) + d];
            Blo[kk].y = P[((size_t)(bi * NC + k0 + 1 ) * DD) + d];
            Bhi[kk].x = P[((size_t)(bi * NC + 16 + k0    ) * DD) + d];
            Bhi[kk].y = P[((size_t)(bi * NC + 16 + k0 + 1) * DD) + d];
        }

        v8f ce = {};  // exclusive scan of low tile
#pragma unroll
        for (int kk = 0; kk < 4; ++kk)
            ce = __builtin_amdgcn_wmma_f32_16x16x4_f32(
                false, La[kk], false, Blo[kk], (short)0, ce, false, false);

        v8f co = {};  // every row = column sums of low tile (carry)
#pragma unroll
        for (int kk = 0; kk < 4; ++kk)
            co = __builtin_amdgcn_wmma_f32_16x16x4_f32(
                false, Oa[kk], false, Blo[kk], (short)0, co, false, false);

        v8f ch = co;  // exclusive scan of high tile + carry
#pragma unroll
        for (int kk = 0; kk < 4; ++kk)
            ch = __builtin_amdgcn_wmma_f32_16x16x4_f32(
                false, La[kk], false, Bhi[kk], (short)0, ch, false, false);

#pragma unroll
        for (int r = 0; r < 8; ++r) {
            const int row = r + 8 * hf;
            O[((size_t)(bi * NC + row     ) * DD) + d] = ce[r];
            O[((size_t)(bi * NC + 16 + row) * DD) + d] = ch[r];
        }
    }
}

// ---------------------------------------------------------------------------
// Pass 3: streaming apply with in-register exclusive scan inside each chunk.
// One thread per (b, chunk, d/4); acc starts at the WMMA-produced chunk base.
// y = x*(w - s*acc_w) + (b - s*acc_b); then acc += grad (exclusive order).
// All float4, lane-contiguous in d -> fully coalesced. 2048 waves.
// ---------------------------------------------------------------------------
__global__ void fw_apply(const float* __restrict__ x,
                         const float* __restrict__ w,
                         const float* __restrict__ bvec,
                         const float* __restrict__ gw,
                         const float* __restrict__ gb,
                         const float* __restrict__ step,
                         const float* __restrict__ bw,
                         const float* __restrict__ bb,
                         float* __restrict__ out) {
    const int blk = blockIdx.x;        // b*NC + c
    const int b   = blk / NC;
    const int c   = blk % NC;
    const int d0  = threadIdx.x * 4;

    const float  s  = step[0];
    const float4 wv = *(const float4*)(w + d0);
    const float4 bv = *(const float4*)(bvec + d0);

    const size_t po = (size_t)blk * DD + d0;
    float4 aw = *(const float4*)(bw + po);
    float4 ab = *(const float4*)(bb + po);

    size_t idx = ((size_t)b * TT + (size_t)c * CT) * DD + d0;

#pragma unroll 4
    for (int t = 0; t < CT; ++t, idx += DD) {
        if (t + 16 < CT) {
            __builtin_prefetch((const char*)(x  + idx + 16 * DD), 0, 0);
            __builtin_prefetch((const char*)(gw + idx + 16 * DD), 0, 0);
            __builtin_prefetch((const char*)(gb + idx + 16 * DD), 0, 0);
        }
        const float4 xv = *(const float4*)(x  + idx);
        const float4 gv = *(const float4*)(gw + idx);
        const float4 hv = *(const float4*)(gb + idx);

        float4 y;
        y.x = __builtin_fmaf(xv.x, wv.x - s * aw.x, bv.x - s * ab.x);
        y.y = __builtin_fmaf(xv.y, wv.y - s * aw.y, bv.y - s * ab.y);
        y.z = __builtin_fmaf(xv.z, wv.z - s * aw.z, bv.z - s * ab.z);
        y.w = __builtin_fmaf(xv.w, wv.w - s * aw.w, bv.w - s * ab.w);

        aw.x += gv.x; aw.y += gv.y; aw.z += gv.z; aw.w += gv.w;
        ab.x += hv.x; ab.y += hv.y; ab.z += hv.z; ab.w += hv.w;

        *(float4*)(out + idx) = y;
    }
}

// ---------------------------------------------------------------------------
extern "C" void kernel_launch(void* const* d_in, const int* in_sizes, int n_in,
                              void* d_out, int out_size, void* d_ws, size_t ws_size,
                              hipStream_t stream) {
    const float* x    = (const float*)d_in[0];
    const float* w    = (const float*)d_in[1];
    const float* bvec = (const float*)d_in[2];
    const float* gw   = (const float*)d_in[3];
    const float* gb   = (const float*)d_in[4];
    const float* step = (const float*)d_in[5];
    float*       out  = (float*)d_out;

    float* pw = (float*)d_ws;          // [BB][NC][DD] chunk partial sums (grad_w)
    float* pb = pw + PSZ;              // [BB][NC][DD] chunk partial sums (grad_b)
    float* bw = pb + PSZ;              // [BB][NC][DD] exclusive chunk bases (w)
    float* bb = bw + PSZ;              // [BB][NC][DD] exclusive chunk bases (b)

    fw_partial_sums<<<dim3(BB * NC), dim3(256), 0, stream>>>(gw, gb, pw, pb);
    fw_scan_partials<<<dim3(BB * (DD / 16)), dim3(32), 0, stream>>>(pw, pb, bw, bb);
    fw_apply<<<dim3(BB * NC), dim3(256), 0, stream>>>(x, w, bvec, gw, gb, step,
                                                      bw, bb, out);
}